// VMFKLLossV2_10376640987794
// MI455X (gfx1250) — compile-verified
//
#include <hip/hip_runtime.h>

// vMF KL divergence, d=64 (v = d/2 = 32), elementwise over kappa.
// - Series branch = constant-coefficient packed Horner in y=x^2 (v_pk_fma_f32),
//   fp32 coefficients underflow to exactly 0 above k=15 -> lossless truncation.
// - Large-v branch in log2 domain: one v_exp_f32; I_33 asymptote derived from
//   I_32 by a constant (ratio is exactly K*x).
// - Large-x branch for all 4 elements in ONE wave-uniform ballot-guarded block
//   (never taken for x<=64).
// - 5 TRANS ops/element on the common path; selects pinned to v_cndmask.

typedef float v2f __attribute__((ext_vector_type(2)));

#define LOG2E_F 1.44269504088896340736f
#define LN2_F   0.69314718055994530942f

__device__ __forceinline__ float f_rcp(float t) { return __builtin_amdgcn_rcpf(t); }
__device__ __forceinline__ float f_rsq(float t) { return __builtin_amdgcn_rsqf(t); }

struct CoefArr { float c[30]; };

// A_k(v) = prod_{j=1..k} 1/(4 j (v+j))  == series coefficients of y^k, y = x^2
__host__ __device__ constexpr CoefArr make_coefs(int V) {
  CoefArr r{};
  double a = 1.0;
  r.c[0] = 1.0f;
  for (int k = 1; k <= 29; ++k) {
    a /= (4.0 * (double)k * (double)(V + k));
    r.c[k] = (float)a;   // fp32: underflows to 0 for large k
  }
  return r;
}

// fp32 coefficient underflow boundary (guaranteed at compile time):
static_assert(make_coefs(32).c[15] != 0.0f && make_coefs(32).c[16] == 0.0f, "v=32 series top");
static_assert(make_coefs(33).c[14] != 0.0f && make_coefs(33).c[15] == 0.0f, "v=33 series top");

// large-x asymptotic pair: e^x / sqrt(2 pi x) * (1 - (4v^2-1)/(8x)), v = 32, 33
__device__ __forceinline__ void lx_pair(float x, float& o32, float& o33) {
  const float rx = f_rcp(x);
  const float bx = __builtin_amdgcn_exp2f(x * LOG2E_F) * f_rsq(6.2831853071795865f * x);
  o32 = bx * fmaf(-511.875f, rx, 1.0f);
  o33 = bx * fmaf(-544.375f, rx, 1.0f);
}

__device__ __forceinline__ float vmf_kl(float x, float lx32, float lx33) {
  constexpr CoefArr C32 = make_coefs(32);
  constexpr CoefArr C33 = make_coefs(33);

  const float l2x = __builtin_amdgcn_logf(x);      // log2(x), shared
  const float y   = x * x;
  const float rx  = f_rcp(x);

  // ---- series branch: packed Horner over y, lane.x = v=32, lane.y = v=33 ----
  const v2f yv = {y, y};
  v2f s = {C32.c[15], 0.0f};                       // k=15 term exists only for v=32
#pragma unroll
  for (int k = 14; k >= 0; --k) {
    const v2f ck = {C32.c[k], C33.c[k]};           // hoisted packed literals
    s = __builtin_elementwise_fma(s, yv, ck);      // -> v_pk_fma_f32
  }
  // (x/2)^32 via 5 squarings, (x/2)^33 with one extra multiply
  const float h = 0.5f * x;
  const float h2 = h * h, h4 = h2 * h2, h8 = h4 * h4, h16 = h8 * h8;
  const float h32p = h16 * h16;
  const float h33p = h32p * h;
  const v2f powv = {h32p, h33p};
  const v2f invf = {3.8003907548547446e-36f,       // 1/32!
                    1.1516335620771953e-37f};      // 1/33!
  const v2f midv = s * powv * invf;

  // ---- large-v asymptotic, log2 domain ----
  // lv32 = corr32 * exp(-0.5 ln(64pi) + 32(1 + ln x - ln 64))
  //      = corr32 * 2^(32*log2(x) - 149.659509)
  float lv32 = __builtin_amdgcn_exp2f(fmaf(l2x, 32.0f, -149.659509f)) * 1.00397715f;
  // lv33/lv32 = (corr33/corr32)*sqrt(32/33)*e*64^32/66^33 * x = K*x, K = e^-4.1898558
  float lv33 = lv32 * x * 0.0151485f;

  // Straight-line branch evaluation (like jnp.where), then cndmask-select.
  float mid32 = midv.x, mid33 = midv.y;
  asm("" : "+v"(mid32), "+v"(mid33), "+v"(lv32), "+v"(lv33));
  const float iv32 = (10.0f * x < 32.0f) ? lv32 : ((x <= 64.0f) ? mid32 : lx32);
  const float iv33 = (10.0f * x < 33.0f) ? lv33 : ((x <= 66.0f) ? mid33 : lx33);

  // ratio = (I_33 + I_32 * 32/x) / I_32 - 32/x   (kept in the reference's
  // cancellation form so fp32 rounding matches the reference)
  const float hd    = 32.0f * rx;
  const float ratio = fmaf(iv32, hd, iv33) * f_rcp(iv32) - hd;

  // kl = x*ratio + 32 ln x - ln I_32 - (lgamma(33) + 32 ln 2), via log2:
  //   32 ln x = (32 ln 2) * log2 x ; ln I_32 = ln2 * log2 I_32
  const float l2iv = __builtin_amdgcn_logf(iv32);
  return fmaf(x, ratio,
         fmaf(l2x, 22.18070977791825f,
         fmaf(l2iv, -LN2_F, -103.73866923403329f)));
}

__global__ __launch_bounds__(256) void vmf_kl_kernel(const float* __restrict__ in,
                                                     float* __restrict__ out, int n) {
  const int n4 = n >> 2;
  const float4* __restrict__ in4 = (const float4*)in;
  float4* __restrict__ out4      = (float4*)out;
  const int stride = (int)(gridDim.x * blockDim.x);
  for (int i = (int)(blockIdx.x * blockDim.x + threadIdx.x); i < n4; i += stride) {
    if (i + stride < n4)
      __builtin_prefetch((const void*)(in4 + i + stride), 0, 1);  // -> global_prefetch_b8
    const float4 xv = in4[i];

    // Single wave-uniform guard for the large-x branch of all 4 elements.
    float ax32 = 0.0f, ax33 = 0.0f, bx32 = 0.0f, bx33 = 0.0f;
    float cx32 = 0.0f, cx33 = 0.0f, dx32 = 0.0f, dx33 = 0.0f;
    const float mx = fmaxf(fmaxf(xv.x, xv.y), fmaxf(xv.z, xv.w));
    if (__builtin_amdgcn_ballot_w32(mx > 64.0f) != 0u) {  // one s_cbranch, normally skipped
      lx_pair(xv.x, ax32, ax33);
      lx_pair(xv.y, bx32, bx33);
      lx_pair(xv.z, cx32, cx33);
      lx_pair(xv.w, dx32, dx33);
    }

    float4 r;
    r.x = vmf_kl(xv.x, ax32, ax33);
    r.y = vmf_kl(xv.y, bx32, bx33);
    r.z = vmf_kl(xv.z, cx32, cx33);
    r.w = vmf_kl(xv.w, dx32, dx33);
    out4[i] = r;
  }
  // tail: n % 4 leftover scalars (none for 4096*8192, but stay general)
  const int base = n4 << 2;
  const int t = (int)(blockIdx.x * blockDim.x + threadIdx.x);
  if (t < (n - base)) {
    const float xt = in[base + t];
    float tx32 = 0.0f, tx33 = 0.0f;
    if (__builtin_amdgcn_ballot_w32(xt > 64.0f) != 0u) lx_pair(xt, tx32, tx33);
    out[base + t] = vmf_kl(xt, tx32, tx33);
  }
}

extern "C" void kernel_launch(void* const* d_in, const int* in_sizes, int n_in,
                              void* d_out, int out_size, void* d_ws, size_t ws_size,
                              hipStream_t stream) {
  (void)n_in; (void)out_size; (void)d_ws; (void)ws_size;
  const float* kappa = (const float*)d_in[0];
  float* out = (float*)d_out;
  const int n  = in_sizes[0];
  const int n4 = n >> 2;

  const int threads = 256;                 // 8 wave32 waves per block
  // ~4 float4 iterations per thread: plenty of waves for every SIMD while
  // amortizing the hoisted coefficient constants, loop overhead, and prefetch.
  long blocks_1to1 = ((long)n4 + threads - 1) / threads;
  int blocks = (int)((blocks_1to1 + 3) / 4);
  if (blocks < 1) blocks = 1;

  vmf_kl_kernel<<<blocks, threads, 0, stream>>>(kappa, out, n);
}